// RGCN_EMB_70368744178403
// MI455X (gfx1250) — compile-verified
//
#include <hip/hip_runtime.h>

#define N_NODES 50000
#define N_RELS 20
#define EMB_DIM 256
#define W_SIZE 16
#define N_CLASSES 8
#define N_EDGES 1600000

#define ROWS_PER_BLOCK 128
#define ELDS_STRIDE 260   // 256 + 4 pad (dwords): lanes hit banks 4*m, conflict-free b64
#define W1T_STRIDE 260    // same padding for the transposed W1 tile

typedef __attribute__((ext_vector_type(2))) float v2f;
typedef __attribute__((ext_vector_type(8))) float v8f;

// ---------------------------------------------------------------------------
// Kernel 1: init. deg = 0, h_agg = 0, out[n][c] = bias2[c]
// ---------------------------------------------------------------------------
__global__ void rgcn_init_kernel(float* __restrict__ deg,
                                 float* __restrict__ hagg,
                                 float* __restrict__ out,
                                 const float* __restrict__ bias2) {
    int i = blockIdx.x * blockDim.x + threadIdx.x;
    if (i < N_RELS * N_NODES)     deg[i]  = 0.0f;
    if (i < N_NODES * W_SIZE)     hagg[i] = 0.0f;
    if (i < N_NODES * N_CLASSES)  out[i]  = bias2[i & 7];
}

// ---------------------------------------------------------------------------
// Kernel 2: deg[(p,s)] += 1 per edge
// ---------------------------------------------------------------------------
__global__ void rgcn_deg_kernel(const int* __restrict__ triples,
                                float* __restrict__ deg) {
    int e = blockIdx.x * blockDim.x + threadIdx.x;
    if (e >= N_EDGES) return;
    int s = triples[3 * e + 0];
    int p = triples[3 * e + 1];
    atomicAdd(&deg[(size_t)p * N_NODES + s], 1.0f);
}

// ---------------------------------------------------------------------------
// Kernel 3: h1[r, m, w] = emb[m, :] @ W1[r]   via V_WMMA_F32_16X16X4_F32
//
// Block = 256 threads (8 waves) owns 128 emb rows, staged ONCE in LDS
// (128 x 260-dword padded tile, 130 KB), then loops all 20 relations:
// stage W1[r] transposed (padded, 16.6 KB), each wave does a 16x16 tile,
// K=256 in steps of 4 (64 WMMAs). Emb is read from HBM exactly once.
// ---------------------------------------------------------------------------
__global__ void __launch_bounds__(256)
rgcn_gemm1_kernel(const float* __restrict__ emb,
                  const float* __restrict__ W1,
                  float* __restrict__ h1) {
    __shared__ float elds[ROWS_PER_BLOCK * ELDS_STRIDE];  // 133,120 B
    __shared__ float w1t[W_SIZE * W1T_STRIDE];            //  16,640 B

    const int m0blk = blockIdx.x * ROWS_PER_BLOCK;
    const int nrows = (N_NODES - m0blk) < ROWS_PER_BLOCK ? (N_NODES - m0blk)
                                                         : ROWS_PER_BLOCK;

    // Stage emb rows once (float4, coalesced within each row)
    for (int idx = threadIdx.x; idx < (nrows << 6); idx += blockDim.x) {
        int row = idx >> 6;            // 0..nrows-1
        int c4  = (idx & 63) << 2;     // 0,4,...,252
        float4 v = *(const float4*)(emb + (size_t)(m0blk + row) * EMB_DIM + c4);
        *(float4*)(&elds[row * ELDS_STRIDE + c4]) = v;
    }

    const int wave  = threadIdx.x >> 5;
    const int lane  = threadIdx.x & 31;
    const int khalf = (lane >> 4) * 2;                 // K pair select
    const bool tile_ok = (wave * 16 + 16) <= nrows;    // wave-uniform

    // A fragment: lane&15 -> M row within wave tile, lane half -> K pair
    const float* arow = &elds[(wave * 16 + (lane & 15)) * ELDS_STRIDE + khalf];
    // B fragment: lane&15 -> N col, contiguous K pair in transposed LDS
    const float* bcol = &w1t[(lane & 15) * W1T_STRIDE + khalf];

    for (int r = 0; r < N_RELS; ++r) {
        __syncthreads();   // emb staged (r==0) / previous relation's reads done
        const float* W1r = W1 + (size_t)r * EMB_DIM * W_SIZE;
        for (int i = threadIdx.x; i < W_SIZE * EMB_DIM; i += blockDim.x) {
            int c  = i >> 8;     // column (0..15)
            int kk = i & 255;    // k
            w1t[c * W1T_STRIDE + kk] = W1r[kk * W_SIZE + c];
        }
        __syncthreads();
        if (!tile_ok) continue;

        v8f acc = {0.f, 0.f, 0.f, 0.f, 0.f, 0.f, 0.f, 0.f};
        #pragma unroll 4
        for (int k = 0; k < EMB_DIM; k += 4) {
            v2f a = *(const v2f*)(arow + k);
            v2f b = *(const v2f*)(bcol + k);
            acc = __builtin_amdgcn_wmma_f32_16x16x4_f32(
                /*neg_a=*/false, a, /*neg_b=*/false, b,
                /*c_mod=*/(short)0, acc, /*reuse_a=*/false, /*reuse_b=*/false);
        }

        // C/D layout: VGPR j -> M = j + 8*(lane>>4), N = lane&15
        float* drow = h1 + ((size_t)r * N_NODES + m0blk + wave * 16
                            + (lane >> 4) * 8) * W_SIZE + (lane & 15);
        #pragma unroll
        for (int j = 0; j < 8; ++j)
            drow[(size_t)j * W_SIZE] = acc[j];
    }
}

// ---------------------------------------------------------------------------
// Kernel 4: h_agg[s, :] += h1[(p,o), :] * (1/deg[(p,s)]) per edge
// h1 (64 MB) is L2-resident (192 MB L2) so the gather mostly hits L2.
// ---------------------------------------------------------------------------
__global__ void rgcn_msg1_kernel(const int* __restrict__ triples,
                                 const float* __restrict__ deg,
                                 const float* __restrict__ h1,
                                 float* __restrict__ hagg) {
    int e = blockIdx.x * blockDim.x + threadIdx.x;
    if (e >= N_EDGES) return;
    int s = triples[3 * e + 0];
    int p = triples[3 * e + 1];
    int o = triples[3 * e + 2];
    float val = 1.0f / deg[(size_t)p * N_NODES + s];
    const float4* hr = (const float4*)(h1 + ((size_t)p * N_NODES + o) * W_SIZE);
    float* dst = hagg + (size_t)s * W_SIZE;
    #pragma unroll
    for (int q = 0; q < 4; ++q) {
        float4 v = hr[q];
        atomicAdd(dst + 4 * q + 0, v.x * val);
        atomicAdd(dst + 4 * q + 1, v.y * val);
        atomicAdd(dst + 4 * q + 2, v.z * val);
        atomicAdd(dst + 4 * q + 3, v.w * val);
    }
}

// ---------------------------------------------------------------------------
// Kernel 5: fused  x = relu(h_agg[o] + b1);  y = x @ W2[p];
//           out[s, :] += y * (1/deg[(p,s)])      (W2, b1 staged in LDS)
// ---------------------------------------------------------------------------
__global__ void rgcn_msg2_kernel(const int* __restrict__ triples,
                                 const float* __restrict__ deg,
                                 const float* __restrict__ hagg,
                                 const float* __restrict__ W2,
                                 const float* __restrict__ bias1,
                                 float* __restrict__ out) {
    __shared__ float w2s[N_RELS * W_SIZE * N_CLASSES];   // 10 KB
    __shared__ float b1s[W_SIZE];
    for (int i = threadIdx.x; i < N_RELS * W_SIZE * N_CLASSES; i += blockDim.x)
        w2s[i] = W2[i];
    if (threadIdx.x < W_SIZE) b1s[threadIdx.x] = bias1[threadIdx.x];
    __syncthreads();

    for (int e = blockIdx.x * blockDim.x + threadIdx.x; e < N_EDGES;
         e += gridDim.x * blockDim.x) {
        int s = triples[3 * e + 0];
        int p = triples[3 * e + 1];
        int o = triples[3 * e + 2];
        float val = 1.0f / deg[(size_t)p * N_NODES + s];
        const float* hb = hagg + (size_t)o * W_SIZE;
        const float* wp = &w2s[p * W_SIZE * N_CLASSES];
        float y[N_CLASSES] = {0.f, 0.f, 0.f, 0.f, 0.f, 0.f, 0.f, 0.f};
        #pragma unroll
        for (int w = 0; w < W_SIZE; ++w) {
            float xv = hb[w] + b1s[w];
            xv = xv > 0.0f ? xv : 0.0f;
            #pragma unroll
            for (int c = 0; c < N_CLASSES; ++c)
                y[c] += xv * wp[w * N_CLASSES + c];
        }
        float* op = out + (size_t)s * N_CLASSES;
        #pragma unroll
        for (int c = 0; c < N_CLASSES; ++c)
            atomicAdd(op + c, y[c] * val);
    }
}

// ---------------------------------------------------------------------------
extern "C" void kernel_launch(void* const* d_in, const int* in_sizes, int n_in,
                              void* d_out, int out_size, void* d_ws, size_t ws_size,
                              hipStream_t stream) {
    const int*   triples = (const int*)d_in[0];
    const float* emb     = (const float*)d_in[1];
    const float* W1      = (const float*)d_in[2];
    const float* W2      = (const float*)d_in[3];
    const float* bias1   = (const float*)d_in[4];
    const float* bias2   = (const float*)d_in[5];
    float* out = (float*)d_out;

    // workspace layout (floats): deg [r*n] | hagg [n*16] | h1 [r*n*16]  ~72 MB
    float* ws   = (float*)d_ws;
    float* deg  = ws;
    float* hagg = deg  + (size_t)N_RELS * N_NODES;
    float* h1   = hagg + (size_t)N_NODES * W_SIZE;

    const int T = 256;

    rgcn_init_kernel<<<(N_RELS * N_NODES + T - 1) / T, T, 0, stream>>>(
        deg, hagg, out, bias2);

    rgcn_deg_kernel<<<(N_EDGES + T - 1) / T, T, 0, stream>>>(triples, deg);

    rgcn_gemm1_kernel<<<(N_NODES + ROWS_PER_BLOCK - 1) / ROWS_PER_BLOCK, T, 0,
                        stream>>>(emb, W1, h1);

    rgcn_msg1_kernel<<<(N_EDGES + T - 1) / T, T, 0, stream>>>(
        triples, deg, h1, hagg);

    rgcn_msg2_kernel<<<1024, T, 0, stream>>>(
        triples, deg, hagg, W2, bias1, out);
}